// SCAN_13975823581397
// MI455X (gfx1250) — compile-verified
//
#include <hip/hip_runtime.h>
#include <math.h>

#define B_   128
#define R_   36
#define RP_  48      // R padded to multiple of 16
#define T_   128
#define D_   1024
#define KC   128     // D chunk staged in LDS
#define SIMG 140     // LDS row strides (pad: stride%64==12, 2*stride%64==24, stride*4%16==0)
#define SCAP 140
#define SSC  140
#define EPSV 1e-8f

typedef __attribute__((ext_vector_type(2))) float v2f;
typedef __attribute__((ext_vector_type(8))) float v8f;
typedef __attribute__((ext_vector_type(4))) int   v4i;

typedef __attribute__((address_space(1))) v4i* gv4i_p;   // global v4i*
typedef __attribute__((address_space(3))) v4i* lv4i_p;   // LDS v4i*

#if defined(__has_builtin)
#if __has_builtin(__builtin_amdgcn_global_load_async_to_lds_b128)
#define USE_ASYNC_LDS 1
#endif
#endif

// Stage 16 bytes global -> LDS. Async path bypasses VGPRs and is tracked by ASYNCcnt.
__device__ __forceinline__ void stage_b128(const float* g, float* l) {
#ifdef USE_ASYNC_LDS
    __builtin_amdgcn_global_load_async_to_lds_b128(
        (gv4i_p)(void*)g,
        (lv4i_p)(void*)l,
        /*imm offset*/0, /*cpol*/0);
#else
    *(float4*)l = *(const float4*)g;
#endif
}

__device__ __forceinline__ void stage_wait() {
#ifdef USE_ASYNC_LDS
#if __has_builtin(__builtin_amdgcn_s_wait_asynccnt)
    __builtin_amdgcn_s_wait_asynccnt(0);
#else
    asm volatile("s_wait_asynccnt 0" ::: "memory");
#endif
#endif
}

__global__ __launch_bounds__(256) void SCAN_13975823581397_kernel(
    const float* __restrict__ img,   // [B, R, D]
    const float* __restrict__ cap,   // [B, T, D]
    float* __restrict__ out)         // [B, T, D]
{
    __shared__ float ls_img[RP_ * SIMG];   // img chunk (rows >= R_ kept zero)
    __shared__ float ls_cap[T_  * SCAP];   // cap chunk
    __shared__ float ls_sc [RP_ * SSC];    // scores -> weights [r][t]
    __shared__ float ls_ni[RP_];
    __shared__ float ls_nc[T_];

    const int b    = blockIdx.x;
    const int tid  = threadIdx.x;
    const int wv   = tid >> 5;      // wave id 0..7
    const int lane = tid & 31;
    const int lh   = lane >> 4;     // lane half: selects K {0,1} vs {2,3} in frags
    const int lm   = lane & 15;

    const float* imgB = img + (size_t)b * R_ * D_;
    const float* capB = cap + (size_t)b * T_ * D_;
    float*       outB = out + (size_t)b * T_ * D_;

    // zero the padded img rows once; staging never touches rows >= R_
    for (int idx = tid; idx < (RP_ - R_) * (KC / 4); idx += 256) {
        int r  = R_ + (idx >> 5);
        int c4 = (idx & 31) << 2;
        *(float4*)(ls_img + r * SIMG + c4) = make_float4(0.f, 0.f, 0.f, 0.f);
    }

    // ---- Phase 1: dot[r][t] = sum_d img[r][d]*cap[t][d], tiled 3x8 of 16x16 ----
    // wave wv owns tiles {wv*3+i}: mt = tile%3 (region rows), nt = tile/3 (token cols)
    // note: all 3 tiles of a wave share nt -> shared B fragment per k-step
    const int t0 = wv * 3;
    const int mt0 = (t0 + 0) % 3, nt0 = (t0 + 0) / 3;
    const int mt1 = (t0 + 1) % 3, nt1 = (t0 + 1) / 3;
    const int mt2 = (t0 + 2) % 3, nt2 = (t0 + 2) / 3;

    v8f acc0 = {}; v8f acc1 = {}; v8f acc2 = {};
    float nacc = 0.0f;   // norm^2 accumulator (img row for tid<36; cap row for 64<=tid<192)

    for (int kc = 0; kc < D_; kc += KC) {
        __syncthreads();  // previous chunk fully consumed
        // stage img chunk: 36 rows x 128 cols (16B per lane-iter)
        for (int idx = tid; idx < R_ * (KC / 4); idx += 256) {
            int r  = idx >> 5;          // KC/4 == 32 float4 per row
            int c4 = (idx & 31) << 2;
            stage_b128(imgB + (size_t)r * D_ + kc + c4, ls_img + r * SIMG + c4);
        }
        // stage cap chunk: 128 rows x 128 cols
        for (int idx = tid; idx < T_ * (KC / 4); idx += 256) {
            int t  = idx >> 5;
            int c4 = (idx & 31) << 2;
            stage_b128(capB + (size_t)t * D_ + kc + c4, ls_cap + t * SCAP + c4);
        }
        stage_wait();
        __syncthreads();

        // norm partial sums from staged rows
        if (tid < R_) {
            const float* row = ls_img + tid * SIMG;
            #pragma unroll 8
            for (int k = 0; k < KC; ++k) nacc = fmaf(row[k], row[k], nacc);
        } else if (tid >= 64 && tid < 64 + T_) {
            const float* row = ls_cap + (tid - 64) * SCAP;
            #pragma unroll 8
            for (int k = 0; k < KC; ++k) nacc = fmaf(row[k], row[k], nacc);
        }

        // WMMA sweep over this K chunk
        #pragma unroll 4
        for (int k0 = 0; k0 < KC; k0 += 4) {
            const int ko = k0 + 2 * lh;  // this lane-half supplies K = ko, ko+1
            {   // tile 0
                v2f a  = *(const v2f*)(ls_img + (mt0 * 16 + lm) * SIMG + ko);
                v2f bb = *(const v2f*)(ls_cap + (nt0 * 16 + lm) * SCAP + ko);
                acc0 = __builtin_amdgcn_wmma_f32_16x16x4_f32(false, a, false, bb,
                                                             (short)0, acc0, false, false);
            }
            {   // tile 1
                v2f a  = *(const v2f*)(ls_img + (mt1 * 16 + lm) * SIMG + ko);
                v2f bb = *(const v2f*)(ls_cap + (nt1 * 16 + lm) * SCAP + ko);
                acc1 = __builtin_amdgcn_wmma_f32_16x16x4_f32(false, a, false, bb,
                                                             (short)0, acc1, false, false);
            }
            {   // tile 2
                v2f a  = *(const v2f*)(ls_img + (mt2 * 16 + lm) * SIMG + ko);
                v2f bb = *(const v2f*)(ls_cap + (nt2 * 16 + lm) * SCAP + ko);
                acc2 = __builtin_amdgcn_wmma_f32_16x16x4_f32(false, a, false, bb,
                                                             (short)0, acc2, false, false);
            }
        }
    }

    // write dot tiles into ls_sc: element (vgpr r, lane) -> M = 8*lh + r, N = lm
    #pragma unroll
    for (int r = 0; r < 8; ++r) {
        ls_sc[(mt0 * 16 + 8 * lh + r) * SSC + nt0 * 16 + lm] = acc0[r];
        ls_sc[(mt1 * 16 + 8 * lh + r) * SSC + nt1 * 16 + lm] = acc1[r];
        ls_sc[(mt2 * 16 + 8 * lh + r) * SSC + nt2 * 16 + lm] = acc2[r];
    }
    if (tid < RP_)                        ls_ni[tid]      = (tid < R_) ? sqrtf(nacc) : 1.0f;
    else if (tid >= 64 && tid < 64 + T_)  ls_nc[tid - 64] = sqrtf(nacc);
    __syncthreads();

    // ---- Phase 2: cosine normalize + softmax over regions (per token column) ----
    if (tid < T_) {
        const int t  = tid;
        const float nct = ls_nc[t];
        float m = -3.402823e38f;
        #pragma unroll 4
        for (int r = 0; r < R_; ++r) {
            float sc = ls_sc[r * SSC + t] / fmaxf(ls_ni[r] * nct, EPSV);
            ls_sc[r * SSC + t] = sc;
            m = fmaxf(m, sc);
        }
        float sum = 0.0f;
        #pragma unroll 4
        for (int r = 0; r < R_; ++r) {
            float e = __expf(ls_sc[r * SSC + t] - m);
            ls_sc[r * SSC + t] = e;
            sum += e;
        }
        const float inv = 1.0f / sum;
        #pragma unroll 4
        for (int r = 0; r < R_; ++r) ls_sc[r * SSC + t] *= inv;
        #pragma unroll
        for (int r = R_; r < RP_; ++r) ls_sc[r * SSC + t] = 0.0f;   // zero K-padding
    }
    __syncthreads();

    // ---- Phase 3: out[t][d] = sum_r w[r][t] * img[r][d];  M=T, N=D chunk, K=R_ ----
    const int mtt = wv * 16;   // wave owns token tile wv
    for (int dc = 0; dc < D_; dc += KC) {
        __syncthreads();
        for (int idx = tid; idx < R_ * (KC / 4); idx += 256) {
            int r  = idx >> 5;
            int c4 = (idx & 31) << 2;
            stage_b128(imgB + (size_t)r * D_ + dc + c4, ls_img + r * SIMG + c4);
        }
        stage_wait();
        __syncthreads();

        #pragma unroll
        for (int nt = 0; nt < KC / 16; ++nt) {
            v8f acc = {};
            #pragma unroll
            for (int k0 = 0; k0 < R_; k0 += 4) {   // K = 36, 9 steps
                const int kr = k0 + 2 * lh;
                v2f a, bb;
                a.x  = ls_sc [ kr      * SSC  + mtt + lm];       // A[m][k] = w[k][t]
                a.y  = ls_sc [(kr + 1) * SSC  + mtt + lm];
                bb.x = ls_img[ kr      * SIMG + nt * 16 + lm];   // B[k][n] = img[k][d]
                bb.y = ls_img[(kr + 1) * SIMG + nt * 16 + lm];
                acc = __builtin_amdgcn_wmma_f32_16x16x4_f32(false, a, false, bb,
                                                            (short)0, acc, false, false);
            }
            #pragma unroll
            for (int r = 0; r < 8; ++r) {
                outB[(size_t)(mtt + 8 * lh + r) * D_ + dc + nt * 16 + lm] = acc[r];
            }
        }
    }
}

extern "C" void kernel_launch(void* const* d_in, const int* in_sizes, int n_in,
                              void* d_out, int out_size, void* d_ws, size_t ws_size,
                              hipStream_t stream) {
    const float* img = (const float*)d_in[0];   // [128, 36, 1024] fp32
    const float* cap = (const float*)d_in[1];   // [128, 128, 1024] fp32
    // d_in[2] = cap_mask (unused by reference forward)
    float* out = (float*)d_out;                 // [128, 128, 1024] fp32
    (void)in_sizes; (void)n_in; (void)out_size; (void)d_ws; (void)ws_size;

    SCAN_13975823581397_kernel<<<dim3(B_), dim3(256), 0, stream>>>(img, cap, out);
}